// Tree_17867063951635
// MI455X (gfx1250) — compile-verified
//
#include <hip/hip_runtime.h>
#include <hip/hip_bf16.h>
#include <cstdint>

#ifndef __has_builtin
#define __has_builtin(x) 0
#endif

// Nodes covering tree levels 0..12. During traversal iteration d, the
// looked-up node index is <= 2^(d+1)-2, so for d <= 12 every lane's index
// is <= 8190 < LDS_N -> the LDS/global split below is wave-uniform.
// 8192 entries = 32 KB LDS (WGP has 320 KB; ~10 blocks/WGP fit), and gives
// an even 16B-per-lane staging loop (8192*4B / (256 thr * 16B) = 8 iters).
#define LDS_N 8192

typedef int v4i __attribute__((ext_vector_type(4)));

typedef __attribute__((address_space(1))) int* gptr_i32;   // b32 variant: plain int*
typedef __attribute__((address_space(3))) int* lptr_i32;
typedef __attribute__((address_space(1))) v4i* gptr_v4i;   // b128 variant: vec4-int*
typedef __attribute__((address_space(3))) v4i* lptr_v4i;

__global__ __launch_bounds__(256) void tree_traverse_kernel(
    const int* __restrict__ x,                  // [n, w] binary features (0/1)
    const int* __restrict__ node_choices,       // [65537]
    const float* __restrict__ node_predictions, // [65537] values in {0.0, 1.0}
    const int* __restrict__ depth_ptr,          // scalar (=15)
    float* __restrict__ out,                    // [n]
    int n, int w)
{
    __shared__ int lds_choices[LDS_N];

    // ---- Stage the hot upper levels of the choice table into LDS ----------
    int* nc_mut = const_cast<int*>(node_choices);  // builtins take non-const as1 ptrs
#if __has_builtin(__builtin_amdgcn_global_load_async_to_lds_b128)
    // 16 B per lane per async op: 8 iterations of full-cacheline DMA into LDS.
    for (int i = threadIdx.x * 4; i < LDS_N; i += blockDim.x * 4) {
        __builtin_amdgcn_global_load_async_to_lds_b128(
            (gptr_v4i)(nc_mut + i), (lptr_v4i)(&lds_choices[i]),
            /*offset=*/0, /*cpol=*/0);
    }
  #if __has_builtin(__builtin_amdgcn_s_wait_asynccnt)
    __builtin_amdgcn_s_wait_asynccnt(0);
  #else
    asm volatile("s_wait_asynccnt 0" ::: "memory");
  #endif
#elif __has_builtin(__builtin_amdgcn_global_load_async_to_lds_b32)
    for (int i = threadIdx.x; i < LDS_N; i += blockDim.x) {
        __builtin_amdgcn_global_load_async_to_lds_b32(
            (gptr_i32)(nc_mut + i), (lptr_i32)(&lds_choices[i]),
            /*offset=*/0, /*cpol=*/0);
    }
  #if __has_builtin(__builtin_amdgcn_s_wait_asynccnt)
    __builtin_amdgcn_s_wait_asynccnt(0);
  #else
    asm volatile("s_wait_asynccnt 0" ::: "memory");
  #endif
#else
    // Fallback: plain load + ds_store path
    for (int i = threadIdx.x; i < LDS_N; i += blockDim.x)
        lds_choices[i] = node_choices[i];
#endif
    __syncthreads();

    const int tid = blockIdx.x * blockDim.x + threadIdx.x;
    if (tid >= n) return;

    const int depth = depth_ptr[0];
    const int* xrow = x + (size_t)tid * (size_t)w;
    __builtin_prefetch(xrow, 0, 0);  // global_prefetch_b8: warm the row head

    int node = 0;
    const int dl = depth < 13 ? depth : 13;

    // Levels 0..12: choice lookup served from LDS (off the HBM latency chain).
    int d = 0;
    for (; d < dl; ++d) {
        const int c   = lds_choices[node];
        const int bit = __builtin_nontemporal_load(xrow + c) & 1;  // th:NT
        node = 2 * node + 1 + bit;
    }
    // Levels 13..14: node index >= 8191, read choice table from L2.
    for (; d < depth; ++d) {
        const int c   = node_choices[node];
        const int bit = __builtin_nontemporal_load(xrow + c) & 1;  // th:NT
        node = 2 * node + 1 + bit;
    }

    // Reference casts prediction to bool; emit exact 0.0/1.0.
    out[tid] = (node_predictions[node] != 0.0f) ? 1.0f : 0.0f;
}

extern "C" void kernel_launch(void* const* d_in, const int* in_sizes, int n_in,
                              void* d_out, int out_size, void* d_ws, size_t ws_size,
                              hipStream_t stream) {
    const int*   x     = (const int*)  d_in[0];  // [N*W] int32
    const int*   nc    = (const int*)  d_in[1];  // [N_NODES] int32
    const float* np    = (const float*)d_in[2];  // [N_NODES] float32
    const int*   depth = (const int*)  d_in[3];  // scalar int32
    float*       out   = (float*)d_out;

    const int n = out_size;                       // 250,000 samples
    const int w = (n > 0) ? (in_sizes[0] / n) : 0; // 784 features

    const int block = 256;                        // 8 wave32s per block
    const int grid  = (n + block - 1) / block;    // ~977 blocks
    tree_traverse_kernel<<<grid, block, 0, stream>>>(x, nc, np, depth, out, n, w);
}